// SphericalHarmonicTransform_66494683677129
// MI455X (gfx1250) — compile-verified
//
#include <hip/hip_runtime.h>
#include <math.h>

typedef __attribute__((ext_vector_type(2))) float v2f;
typedef __attribute__((ext_vector_type(8))) float v8f;

#define NBLK 1024
#define BLOCK 128
#define WAVES 4                  // 128 threads / wave32
#define FSTRIDE 34               // padded point-stride for feature rows (bank-conflict-free)
#define FEAT_FLOATS (80 * FSTRIDE)        // 80 feature rows (72 used + 8 zero pad)
#define RP_FLOATS (32 * 16)               // r^l per point, padded to 16
#define WAVE_FLOATS (FEAT_FLOATS + RP_FLOATS)   // 3232 floats per wave
#define MOM_ELEMS (80 * 16)               // moment matrix in d_ws

__global__ void sh_zero_moments(float* mom) {
    int i = blockIdx.x * blockDim.x + threadIdx.x;
    if (i < MOM_ELEMS) mom[i] = 0.0f;
}

// Main kernel: moments[row][l] += sum_i feat_row(i) * mask*r^l via f32 WMMA,
// K dimension = point index (4 points per v_wmma_f32_16x16x4_f32).
__global__ void __launch_bounds__(BLOCK) sh_moments(const float* __restrict__ pos,
                                                    long long npts,
                                                    float* __restrict__ mom) {
    __shared__ float lds[WAVES * WAVE_FLOATS];   // 51712 bytes

    const int lane = threadIdx.x & 31;
    const int w    = threadIdx.x >> 5;
    const int half = lane >> 4;       // 0: lanes 0-15, 1: lanes 16-31
    const int l16  = lane & 15;

    float* fb = &lds[w * WAVE_FLOATS];            // feature rows [80][FSTRIDE]
    float* rp = fb + FEAT_FLOATS;                 // r-powers     [32][16]

    // Zero the 8 pad feature rows once (each lane clears its own column).
#pragma unroll
    for (int r = 0; r < 8; ++r) fb[(72 + r) * FSTRIDE + lane] = 0.0f;

    v8f acc[5] = {};   // 5 tiles of 16 rows x 16 l-columns, f32 accum

    const long long wid    = (long long)blockIdx.x * WAVES + w;
    const long long wstep  = (long long)NBLK * WAVES * 32;

    for (long long base = wid * 32; base < npts; base += wstep) {
        long long idx = base + lane;

        // ---- per-lane point load + feature computation ----
        float x = 0.0f, y = 0.0f, zz = 0.0f;
        if (idx < npts) {
            const float* p = pos + idx * 3;
            x = p[0]; y = p[1]; zz = p[2];
        }
        if (idx + wstep < npts)
            __builtin_prefetch(pos + (idx + wstep) * 3, 0, 1);  // global_prefetch_b8

        // complex power tables (z1 = -x/2 - i*y/2, z2 = x/2 - i*y/2)
        float p1r[11], p1i[11], p2r[11], p2i[11];
        p1r[0] = 1.0f; p1i[0] = 0.0f; p2r[0] = 1.0f; p2i[0] = 0.0f;
        const float z1r = -0.5f * x, z1i = -0.5f * y;
        const float z2r =  0.5f * x, z2i = -0.5f * y;
#pragma unroll
        for (int n = 1; n <= 10; ++n) {
            p1r[n] = z1r * p1r[n - 1] - z1i * p1i[n - 1];
            p1i[n] = z1r * p1i[n - 1] + z1i * p1r[n - 1];
            p2r[n] = z2r * p2r[n - 1] - z2i * p2i[n - 1];
            p2i[n] = z2r * p2i[n - 1] + z2i * p2r[n - 1];
        }

        // 36 (p,q) pairs, q<=p, p+q<=10; rows 0..35 = zre*z, rows 36..71 = zim*z
        const int off[6] = {0, 11, 20, 27, 32, 35};
#pragma unroll
        for (int q = 0; q <= 5; ++q) {
#pragma unroll
            for (int p = q; p <= 10 - q; ++p) {
                int row = off[q] + (p - q);
                float zre = p1r[p] * p1r[p] - p2i[q] * p2i[q];   // matches reference
                float zim = p1r[p] * p2i[q] + p1i[p] * p2r[q];
                fb[row * FSTRIDE + lane]        = zre * zz;
                fb[(36 + row) * FSTRIDE + lane] = zim * zz;
            }
        }

        // masked r^l (mask folded in: rl[0] = (r>0), so l=0 term is masked too)
        float r  = sqrtf(x * x + y * y + zz * zz);
        float rl = (r > 0.0f) ? 1.0f : 0.0f;
#pragma unroll
        for (int l = 0; l < 11; ++l) { rp[lane * 16 + l] = rl; rl *= r; }
#pragma unroll
        for (int l = 11; l < 16; ++l) rp[lane * 16 + l] = 0.0f;

        // ---- WMMA accumulation: 8 K-steps of 4 points, 5 row tiles ----
        // A 16x4 layout: lane -> M = l16; VGPR0 = K(2*half), VGPR1 = K(2*half+1)
        // B  4x16 layout: lane -> N = l16; VGPR0 = K(2*half), VGPR1 = K(2*half+1)
#pragma unroll
        for (int t = 0; t < 8; ++t) {
            v2f b;
            b.x = rp[(4 * t + 2 * half + 0) * 16 + l16];
            b.y = rp[(4 * t + 2 * half + 1) * 16 + l16];
#pragma unroll
            for (int T = 0; T < 5; ++T) {
                int ab = (T * 16 + l16) * FSTRIDE + 4 * t + 2 * half;
                v2f a;
                a.x = fb[ab];
                a.y = fb[ab + 1];
                acc[T] = __builtin_amdgcn_wmma_f32_16x16x4_f32(
                    false, a, false, b, (short)0, acc[T], false, false);
            }
        }
    }

    // ---- block reduction (4 waves -> 1), then global f32 atomics ----
    __syncthreads();
#pragma unroll
    for (int T = 0; T < 5; ++T)
#pragma unroll
        for (int r = 0; r < 8; ++r)
            lds[w * 1280 + T * 256 + r * 32 + lane] = acc[T][r];
    __syncthreads();

    for (int j = threadIdx.x; j < 1280; j += BLOCK) {
        float s = 0.0f;
#pragma unroll
        for (int w2 = 0; w2 < WAVES; ++w2) s += lds[w2 * 1280 + j];
        int T = j >> 8, rem = j & 255, r = rem >> 5, ln = rem & 31;
        // C/D layout: VGPR r -> M = r + 8*(ln>=16), N = ln&15
        int Mrow = T * 16 + r + ((ln >> 4) << 3);
        int N    = ln & 15;
        atomicAdd(&mom[Mrow * 16 + N], s);
    }
}

// Tiny finalize: apply 1/(p!q!s!) coefficients + scales to form the 121 outputs.
__global__ void sh_finalize(const float* __restrict__ mom, float* __restrict__ out,
                            int out_size) {
    int j = threadIdx.x;
    if (j >= 121 || j >= out_size) return;

    int l = 0;
    while ((l + 1) * (l + 1) <= j) ++l;
    int k = j - l * l;
    int M = k - l;
    int m = (M < 0) ? -M : M;

    double fact[21];
    fact[0] = 1.0;
    for (int i = 1; i <= 20; ++i) fact[i] = fact[i - 1] * (double)i;

    const int off[6] = {0, 11, 20, 27, 32, 35};
    double sr = 0.0, si = 0.0;
    for (int p = m; p <= 10; ++p) {
        int q = p - m;
        int s = l - p - q;
        if (q >= 0 && s >= 0) {
            int row = off[q] + (p - q);
            double c = 1.0 / (fact[p] * fact[q] * fact[s]);
            sr += c * (double)mom[row * 16 + l];
            si += c * (double)mom[(36 + row) * 16 + l];
        }
    }
    double scale = sqrt(fact[l + m] * fact[l - m]) *
                   sqrt((2.0 * l + 1.0) / (4.0 * M_PI));
    double v;
    if (M == 0) {
        v = sr * scale;
    } else {
        double f = sqrt(2.0) * ((m & 1) ? -1.0 : 1.0);
        v = (M < 0 ? si : sr) * f * scale;
    }
    out[j] = (float)v;
}

extern "C" void kernel_launch(void* const* d_in, const int* in_sizes, int n_in,
                              void* d_out, int out_size, void* d_ws, size_t ws_size,
                              hipStream_t stream) {
    const float* pos = (const float*)d_in[0];
    long long npts   = (long long)in_sizes[0] / 3;
    float* out       = (float*)d_out;
    float* mom       = (float*)d_ws;   // 80*16 f32 moment accumulator

    sh_zero_moments<<<dim3((MOM_ELEMS + 255) / 256), dim3(256), 0, stream>>>(mom);
    sh_moments<<<dim3(NBLK), dim3(BLOCK), 0, stream>>>(pos, npts, mom);
    sh_finalize<<<dim3(1), dim3(128), 0, stream>>>(mom, out, out_size);
}